// REINFORCE_33792802685719
// MI455X (gfx1250) — compile-verified
//
#include <hip/hip_runtime.h>

// Problem shape (fixed by the reference)
static constexpr int B = 4096;
static constexpr int T = 2048;
static constexpr float GAMMA = 0.99f;
static constexpr float CLIPV = 5.0f;
static constexpr int CH = 8;          // elements per thread in the scan
static constexpr int SCAN_THREADS = T / CH;   // 256
// gamma^8, precomputed in double then truncated
static constexpr float Q8 = 0.92274469442792012695f;

typedef __attribute__((ext_vector_type(2))) float v2f;
typedef __attribute__((ext_vector_type(8))) float v8f;

__device__ __forceinline__ float log_sigmoid(float x) {
    // stable: min(x,0) - log1p(exp(-|x|))
    float mn = fminf(x, 0.0f);
    return mn - log1pf(expf(-fabsf(x)));
}

// ---------------------------------------------------------------------------
// K1: per-row reverse discounted scan.
// One block per row b (4096 blocks x 256 threads). Each thread owns 8
// contiguous t's (float4 x2 loads), does a serial local suffix scan, then the
// 256 chunk totals are combined with a log-step weighted suffix scan in LDS
// (ratio Q8 = gamma^8).  cum is written to d_out + T.
// ---------------------------------------------------------------------------
__global__ void __launch_bounds__(SCAN_THREADS)
reinforce_scan_kernel(const float* __restrict__ logits,
                      const float* __restrict__ weight,
                      float* __restrict__ cum) {
    const int b = blockIdx.x;
    const int j = threadIdx.x;                 // chunk index 0..255
    const size_t base = (size_t)b * T + (size_t)j * CH;

    // wide loads of logits & weight
    float4 lg0 = *reinterpret_cast<const float4*>(logits + base);
    float4 lg1 = *reinterpret_cast<const float4*>(logits + base + 4);
    float4 w0  = *reinterpret_cast<const float4*>(weight + base);
    float4 w1  = *reinterpret_cast<const float4*>(weight + base + 4);

    float wr[CH];
    wr[0] = w0.x * log_sigmoid(lg0.x);
    wr[1] = w0.y * log_sigmoid(lg0.y);
    wr[2] = w0.z * log_sigmoid(lg0.z);
    wr[3] = w0.w * log_sigmoid(lg0.w);
    wr[4] = w1.x * log_sigmoid(lg1.x);
    wr[5] = w1.y * log_sigmoid(lg1.y);
    wr[6] = w1.z * log_sigmoid(lg1.z);
    wr[7] = w1.w * log_sigmoid(lg1.w);

    // local reverse scan within the chunk
    float c[CH];
    c[CH - 1] = wr[CH - 1];
#pragma unroll
    for (int i = CH - 2; i >= 0; --i) c[i] = wr[i] + GAMMA * c[i + 1];

    // log-step weighted suffix scan over chunk totals (ping-pong LDS)
    __shared__ float s[2][SCAN_THREADS];
    s[0][j] = c[0];
    int src = 0;
    float qp = Q8;                 // Q8^off for off = 1,2,4,...
    for (int off = 1; off < SCAN_THREADS; off <<= 1) {
        __syncthreads();
        float v = s[src][j];
        if (j + off < SCAN_THREADS) v += qp * s[src][j + off];
        s[1 - src][j] = v;
        src = 1 - src;
        qp *= qp;
    }
    __syncthreads();
    // carry entering this chunk from the right = inclusive scan of chunk j+1
    float R = (j + 1 < SCAN_THREADS) ? s[src][j + 1] : 0.0f;

    // powers gamma^1..gamma^8
    float gp[CH + 1];
    gp[0] = 1.0f;
#pragma unroll
    for (int k = 1; k <= CH; ++k) gp[k] = gp[k - 1] * GAMMA;
#pragma unroll
    for (int i = 0; i < CH; ++i) c[i] += gp[CH - i] * R;

    float4 o0 = make_float4(c[0], c[1], c[2], c[3]);
    float4 o1 = make_float4(c[4], c[5], c[6], c[7]);
    *reinterpret_cast<float4*>(cum + base)     = o0;
    *reinterpret_cast<float4*>(cum + base + 4) = o1;
}

// ---------------------------------------------------------------------------
// K2: partial column sums of (cum - baseline) over a chunk of rows, computed
// with V_WMMA_F32_16X16X4_F32 against an all-ones A matrix.  Each wave owns
// 16 t-columns; each fragment is 4 batch rows x 16 columns.  With A == ones
// the k-lane permutation of the B fragment cancels out; only the N = lane&15
// mapping (documented for C/D) matters.
// Manually unrolled: 16 rows (4 WMMA) per iteration, all loads as immediate
// offsets (row stride 4*T*4 = 32 KiB, max offset 13*T*4 = 0x1A000 < 2^23)
// off two base pointers that advance by one constant per iteration.
// ---------------------------------------------------------------------------
__global__ void __launch_bounds__(256)
reinforce_colsum_wmma_kernel(const float* __restrict__ cum,
                             const float* __restrict__ baselines,
                             float* __restrict__ pws,   // [nchunk][T]
                             int rowsPerChunk) {
    const int lane = threadIdx.x & 31;
    const int wave = threadIdx.x >> 5;
    const int t0 = blockIdx.x * 128 + wave * 16;   // 8 waves * 16 cols = 128
    const int col = t0 + (lane & 15);
    const int khalf = (lane >> 4) << 1;            // 0 or 2
    const int rbase = blockIdx.y * rowsPerChunk;

    const float* pc = cum       + (size_t)(rbase + khalf) * T + col;
    const float* pb = baselines + (size_t)(rbase + khalf) * T + col;

    v2f a; a.x = 1.0f; a.y = 1.0f;                 // ones A (16x4)
    v8f acc = {};
    const int iters = rowsPerChunk >> 4;           // 16 rows / iteration
    for (int it = 0; it < iters; ++it) {
        v2f b0, b1, b2, b3;
        b0.x = pc[0 * T]  - pb[0 * T];   b0.y = pc[1 * T]  - pb[1 * T];
        b1.x = pc[4 * T]  - pb[4 * T];   b1.y = pc[5 * T]  - pb[5 * T];
        b2.x = pc[8 * T]  - pb[8 * T];   b2.y = pc[9 * T]  - pb[9 * T];
        b3.x = pc[12 * T] - pb[12 * T];  b3.y = pc[13 * T] - pb[13 * T];
        acc = __builtin_amdgcn_wmma_f32_16x16x4_f32(false, a, false, b0, (short)0, acc, false, false);
        acc = __builtin_amdgcn_wmma_f32_16x16x4_f32(false, a, false, b1, (short)0, acc, false, false);
        acc = __builtin_amdgcn_wmma_f32_16x16x4_f32(false, a, false, b2, (short)0, acc, false, false);
        acc = __builtin_amdgcn_wmma_f32_16x16x4_f32(false, a, false, b3, (short)0, acc, false, false);
        pc += 16 * T;
        pb += 16 * T;
    }
    // D row M=0 lives in acc[0], lanes 0..15 hold N=0..15
    if (lane < 16) pws[(size_t)blockIdx.y * T + t0 + lane] = acc[0];
}

// K2b: reduce chunk partials -> mean[t]
__global__ void __launch_bounds__(256)
reinforce_mean_kernel(const float* __restrict__ pws, float* __restrict__ mean,
                      int nchunk) {
    const int t = blockIdx.x * 256 + threadIdx.x;
    float s = 0.0f;
    for (int c = 0; c < nchunk; ++c) s += pws[(size_t)c * T + t];
    mean[t] = s * (1.0f / (float)B);
}

// ---------------------------------------------------------------------------
// K3: partial column sums of clip(cum - baseline - mean) * log_probs, again
// via the ones-A fp32 WMMA column reduction with the same unrolled,
// immediate-offset load structure.
// ---------------------------------------------------------------------------
__global__ void __launch_bounds__(256)
reinforce_obj_wmma_kernel(const float* __restrict__ cum,
                          const float* __restrict__ baselines,
                          const float* __restrict__ log_probs,
                          const float* __restrict__ mean,
                          float* __restrict__ pws,   // [nchunk][T]
                          int rowsPerChunk) {
    const int lane = threadIdx.x & 31;
    const int wave = threadIdx.x >> 5;
    const int t0 = blockIdx.x * 128 + wave * 16;
    const int col = t0 + (lane & 15);
    const int khalf = (lane >> 4) << 1;
    const int rbase = blockIdx.y * rowsPerChunk;
    const float m = mean[col];

    const float* pc = cum       + (size_t)(rbase + khalf) * T + col;
    const float* pb = baselines + (size_t)(rbase + khalf) * T + col;
    const float* pl = log_probs + (size_t)(rbase + khalf) * T + col;

    v2f a; a.x = 1.0f; a.y = 1.0f;
    v8f acc = {};
    const int iters = rowsPerChunk >> 4;           // 16 rows / iteration
    for (int it = 0; it < iters; ++it) {
#pragma unroll
        for (int u = 0; u < 4; ++u) {
            const int o0 = (u * 4) * T;
            const int o1 = (u * 4 + 1) * T;
            float a0 = pc[o0] - pb[o0] - m;
            float a1 = pc[o1] - pb[o1] - m;
            a0 = fminf(fmaxf(a0, -CLIPV), CLIPV);
            a1 = fminf(fmaxf(a1, -CLIPV), CLIPV);
            v2f bf;
            bf.x = a0 * pl[o0];
            bf.y = a1 * pl[o1];
            acc = __builtin_amdgcn_wmma_f32_16x16x4_f32(
                false, a, false, bf, (short)0, acc, false, false);
        }
        pc += 16 * T;
        pb += 16 * T;
        pl += 16 * T;
    }
    if (lane < 16) pws[(size_t)blockIdx.y * T + t0 + lane] = acc[0];
}

// K3b: reduce chunk partials -> objective[t]
__global__ void __launch_bounds__(256)
reinforce_objreduce_kernel(const float* __restrict__ pws,
                           float* __restrict__ obj, int nchunk) {
    const int t = blockIdx.x * 256 + threadIdx.x;
    float s = 0.0f;
    for (int c = 0; c < nchunk; ++c) s += pws[(size_t)c * T + t];
    obj[t] = s;
}

extern "C" void kernel_launch(void* const* d_in, const int* in_sizes, int n_in,
                              void* d_out, int out_size, void* d_ws, size_t ws_size,
                              hipStream_t stream) {
    (void)in_sizes; (void)n_in; (void)out_size;
    const float* log_probs = (const float*)d_in[0];   // [B,T]
    const float* logits    = (const float*)d_in[1];   // [B,T,1]
    const float* weight    = (const float*)d_in[2];   // [B,T]
    const float* baselines = (const float*)d_in[3];   // [B,T,1]

    float* out = (float*)d_out;        // [0..T): objective, [T..): cum rewards
    float* cum = out + T;
    float* ws  = (float*)d_ws;
    float* mean = ws;                  // T floats
    float* pws  = ws + T;              // nchunk * T floats

    // pick largest power-of-two chunk count that fits in the workspace
    size_t wsFloats = ws_size / sizeof(float);
    int nchunk = 32;
    while (nchunk > 1 && (size_t)(nchunk + 1) * T > wsFloats) nchunk >>= 1;
    int rowsPerChunk = B / nchunk;     // divisible by 16 for all choices

    // K1: reverse discounted scan -> cum
    reinforce_scan_kernel<<<B, SCAN_THREADS, 0, stream>>>(logits, weight, cum);

    // K2: WMMA column partial sums of (cum - baseline)
    dim3 gRed(T / 128, nchunk);
    reinforce_colsum_wmma_kernel<<<gRed, 256, 0, stream>>>(cum, baselines, pws,
                                                           rowsPerChunk);
    // K2b: partials -> mean
    reinforce_mean_kernel<<<T / 256, 256, 0, stream>>>(pws, mean, nchunk);

    // K3: WMMA column partial sums of clip(adv)*log_probs
    reinforce_obj_wmma_kernel<<<gRed, 256, 0, stream>>>(cum, baselines, log_probs,
                                                        mean, pws, rowsPerChunk);
    // K3b: partials -> objective
    reinforce_objreduce_kernel<<<T / 256, 256, 0, stream>>>(pws, out, nchunk);
}